// SelectiveSSM_27127013441892
// MI455X (gfx1250) — compile-verified
//
#include <hip/hip_runtime.h>
#include <math.h>

// ---------------- problem constants (from reference) ----------------
#define DMODEL   1024
#define DSTATE   16
#define DCONV    4
#define DINNER   2048          // D_MODEL * EXPAND
#define BATCH    2
#define SEQ      2048
#define NTOK     (BATCH*SEQ)   // 4096 tokens

typedef __attribute__((ext_vector_type(2))) float v2f;
typedef __attribute__((ext_vector_type(8))) float v8f;
typedef __attribute__((ext_vector_type(4))) int   v4i;

// CDNA5 async global->LDS (ASYNCcnt-tracked), with aligned-float4 fallback
#if defined(__has_builtin)
#if __has_builtin(__builtin_amdgcn_global_load_async_to_lds_b128)
#define HAVE_ASYNC_LDS 1
#endif
#endif
#ifndef HAVE_ASYNC_LDS
#define HAVE_ASYNC_LDS 0
#endif

#if HAVE_ASYNC_LDS
typedef __attribute__((address_space(1))) v4i gas_v4i;   // global
typedef __attribute__((address_space(3))) v4i las_v4i;   // LDS
#endif

// =====================================================================
// f32 GEMM via V_WMMA_F32_16X16X4_F32 (exact f32, matrix pipe)
// C[M,N] = A[M,K] @ B[K,N], row-major, M%128==0, N%64==0, K%16==0
// Block = 256 threads (8 waves). Block tile 128x64, wave tile 32x32.
// Double-buffered LDS, async global->LDS b128 copies.
// =====================================================================
constexpr int BM = 128, BN = 64, KB = 16;
constexpr int ASTR = KB + 4;   // 20 floats (80B): 16B-aligned rows, 16 distinct banks
constexpr int BSTR = BN + 4;   // 68 floats (272B): 16B-aligned rows
constexpr int ATILE = BM * ASTR;
constexpr int BTILE = KB * BSTR;

__device__ __forceinline__ void cp_g2l_b128(const float* gp, float* lp)
{
#if HAVE_ASYNC_LDS
    __builtin_amdgcn_global_load_async_to_lds_b128(
        (gas_v4i*)gp, (las_v4i*)lp, 0, 0);
#else
    *(float4*)lp = *(const float4*)gp;
#endif
}

__device__ __forceinline__ void wait_async_lds()
{
#if HAVE_ASYNC_LDS
    asm volatile("s_wait_asynccnt 0" ::: "memory");
#endif
}

__global__ __launch_bounds__(256)
void gemm_f32_wmma(const float* __restrict__ A, const float* __restrict__ B,
                   float* __restrict__ C, int M, int N, int K)
{
    __shared__ __attribute__((aligned(16))) float As[2 * ATILE];  // 20 KB
    __shared__ __attribute__((aligned(16))) float Bs[2 * BTILE];  //  8.7 KB

    const int tid  = threadIdx.x;
    const int lane = tid & 31;
    const int wave = tid >> 5;
    const int half = lane >> 4;     // 0: lanes 0-15, 1: lanes 16-31
    const int lr   = lane & 15;
    const int wm   = (wave >> 1) * 32;   // 0,32,64,96
    const int wn   = (wave & 1) * 32;    // 0,32
    const int row0 = blockIdx.y * BM;
    const int col0 = blockIdx.x * BN;

    // stage one 128x16 A tile + 16x64 B tile into buffer `buf` (b128 per lane)
    auto stage = [&](int buf, int kb) {
        // A: 512 float4s, 2 per thread
        #pragma unroll
        for (int i = 0; i < 2; ++i) {
            const int idx = tid + i * 256;
            const int m = idx >> 2, kq = (idx & 3) * 4;
            cp_g2l_b128(&A[(size_t)(row0 + m) * K + kb + kq],
                        &As[buf * ATILE + m * ASTR + kq]);
        }
        // B: 256 float4s, 1 per thread
        {
            const int k = tid >> 4, nq = (tid & 15) * 4;
            cp_g2l_b128(&B[(size_t)(kb + k) * N + col0 + nq],
                        &Bs[buf * BTILE + k * BSTR + nq]);
        }
    };

    v8f acc[2][2];
    for (int mi = 0; mi < 2; ++mi)
        for (int ni = 0; ni < 2; ++ni)
            for (int r = 0; r < 8; ++r) acc[mi][ni][r] = 0.0f;

    stage(0, 0);
    int buf = 0;
    for (int kb = 0; kb < K; kb += KB) {
        wait_async_lds();          // tile `kb` resident in LDS (ASYNCcnt)
        __syncthreads();
        if (kb + KB < K) stage(buf ^ 1, kb + KB);   // overlap next copy w/ compute

        const float* as = &As[buf * ATILE];
        const float* bs = &Bs[buf * BTILE];

        // 4 WMMA k-steps of 4 each; 16 v_wmma per K-tile per wave
        #pragma unroll
        for (int k0 = 0; k0 < KB; k0 += 4) {
            const int kk = k0 + 2 * half;  // A/B frag: lanes<16 get k0,k0+1; lanes>=16 get k0+2,k0+3
            v2f af[2], bf[2];
            #pragma unroll
            for (int mi = 0; mi < 2; ++mi) {
                const int m = wm + mi * 16 + lr;
                af[mi].x = as[m * ASTR + kk];
                af[mi].y = as[m * ASTR + kk + 1];
            }
            #pragma unroll
            for (int ni = 0; ni < 2; ++ni) {
                const int n = wn + ni * 16 + lr;
                bf[ni].x = bs[kk * BSTR + n];
                bf[ni].y = bs[(kk + 1) * BSTR + n];
            }
            #pragma unroll
            for (int mi = 0; mi < 2; ++mi)
                #pragma unroll
                for (int ni = 0; ni < 2; ++ni)
                    acc[mi][ni] = __builtin_amdgcn_wmma_f32_16x16x4_f32(
                        false, af[mi], false, bf[ni],
                        (short)0, acc[mi][ni], false, false);
        }
        __syncthreads();   // everyone done reading `buf` before it is re-staged
        buf ^= 1;
    }

    // store: C/D layout -> VGPR r holds M=r (lanes 0-15) / M=8+r (lanes 16-31)
    #pragma unroll
    for (int mi = 0; mi < 2; ++mi)
        #pragma unroll
        for (int ni = 0; ni < 2; ++ni) {
            const int col = col0 + wn + ni * 16 + lr;
            #pragma unroll
            for (int r = 0; r < 8; ++r) {
                const int row = row0 + wm + mi * 16 + r + 8 * half;
                C[(size_t)row * N + col] = acc[mi][ni][r];
            }
        }
}

// =====================================================================
// Depthwise causal conv (width 4) + bias + SiLU.
// x_inner = xr[:, :, 0:DINNER]  (xr row width = 2*DINNER)
// =====================================================================
__global__ __launch_bounds__(256)
void conv_silu_kernel(const float* __restrict__ xr, const float* __restrict__ cw,
                      const float* __restrict__ cb, float* __restrict__ xc)
{
    const size_t g = (size_t)blockIdx.x * 256 + threadIdx.x;
    if (g >= (size_t)NTOK * DINNER) return;
    const int d = (int)(g % DINNER);
    const int t = (int)((g / DINNER) % SEQ);
    const int b = (int)(g / ((size_t)DINNER * SEQ));

    float acc = cb[d];
    #pragma unroll
    for (int j = 0; j < DCONV; ++j) {
        const int tt = t - (DCONV - 1) + j;
        if (tt >= 0)
            acc += cw[d * DCONV + j] *
                   xr[((size_t)(b * SEQ + tt)) * (2 * DINNER) + d];
    }
    const float s = acc * (1.0f / (1.0f + __expf(-acc)));   // SiLU
    xc[((size_t)(b * SEQ + t)) * DINNER + d] = s;
}

// =====================================================================
// ssm_params = xc @ W_x  (K=2048, N=33). One block per token; xc row
// staged in LDS; threads 0..32 each own one output column.
// delta=softplus(col 0), B=cols 1..16, C=cols 17..32.
// =====================================================================
__global__ __launch_bounds__(256)
void ssm_params_kernel(const float* __restrict__ xc, const float* __restrict__ Wx,
                       float* __restrict__ delta, float* __restrict__ Bc,
                       float* __restrict__ Cc)
{
    __shared__ float sx[DINNER];   // 8 KB
    const int tok = blockIdx.x;
    const int tid = threadIdx.x;
    for (int i = tid; i < DINNER; i += 256)
        sx[i] = xc[(size_t)tok * DINNER + i];
    __syncthreads();

    if (tid < 33) {
        float acc = 0.f;
        for (int k = 0; k < DINNER; ++k)
            acc = fmaf(sx[k], Wx[k * 33 + tid], acc);
        if (tid == 0) {
            delta[tok] = (acc > 20.f) ? acc : log1pf(__expf(acc));  // softplus
        } else if (tid < 17) {
            Bc[(size_t)tok * DSTATE + (tid - 1)] = acc;
        } else {
            Cc[(size_t)tok * DSTATE + (tid - 17)] = acc;
        }
    }
}

// =====================================================================
// Selective scan. One thread per (b,d); h[16] in VGPRs; per-token
// dt/B/C staged in LDS in 64-step chunks (broadcast reads).
// =====================================================================
__global__ __launch_bounds__(256)
void scan_kernel(const float* __restrict__ xc, const float* __restrict__ delta,
                 const float* __restrict__ Bc, const float* __restrict__ Cc,
                 const float* __restrict__ A_log, float* __restrict__ y)
{
    const int tid = threadIdx.x;
    const int b   = blockIdx.x >> 3;                 // 8 blocks per batch
    const int d   = ((blockIdx.x & 7) << 8) + tid;   // 0..2047

    __shared__ float sdt[64];
    __shared__ float sB[64 * DSTATE];
    __shared__ float sC[64 * DSTATE];

    float Ad[DSTATE], h[DSTATE];
    #pragma unroll
    for (int n = 0; n < DSTATE; ++n) {
        Ad[n] = -__expf(A_log[(size_t)d * DSTATE + n]);
        h[n]  = 0.f;
    }

    for (int tc = 0; tc < SEQ; tc += 64) {
        for (int i = tid; i < 64; i += 256)
            sdt[i] = delta[b * SEQ + tc + i];
        for (int i = tid; i < 64 * DSTATE; i += 256) {
            const size_t base = (size_t)(b * SEQ + tc) * DSTATE + i;
            sB[i] = Bc[base];
            sC[i] = Cc[base];
        }
        __syncthreads();

        for (int tt = 0; tt < 64; ++tt) {
            const size_t tok = (size_t)(b * SEQ + tc + tt);
            const float dt = sdt[tt];
            const float xv = xc[tok * DINNER + d];   // coalesced
            float yv = 0.f;
            #pragma unroll
            for (int n = 0; n < DSTATE; ++n) {
                const float ab = __expf(Ad[n] * dt);           // v_exp_f32 (TRANS)
                h[n] = fmaf(ab, h[n], dt * sB[tt * DSTATE + n] * xv);
                yv   = fmaf(sC[tt * DSTATE + n], h[n], yv);
            }
            y[tok * DINNER + d] = yv;                 // coalesced
        }
        __syncthreads();
    }
}

// =====================================================================
// Epilogue: y = (y + xc*D) * silu(res), res = xr[:, :, DINNER:2*DINNER]
// =====================================================================
__global__ __launch_bounds__(256)
void epilogue_kernel(float* __restrict__ y, const float* __restrict__ xc,
                     const float* __restrict__ xr, const float* __restrict__ Dp)
{
    const size_t g = (size_t)blockIdx.x * 256 + threadIdx.x;
    if (g >= (size_t)NTOK * DINNER) return;
    const int d = (int)(g % DINNER);
    const size_t tok = g / DINNER;
    const float res = xr[tok * (2 * DINNER) + DINNER + d];
    const float sr  = res * (1.0f / (1.0f + __expf(-res)));
    y[g] = (y[g] + xc[g] * Dp[d]) * sr;
}

// =====================================================================
// launch
// =====================================================================
extern "C" void kernel_launch(void* const* d_in, const int* in_sizes, int n_in,
                              void* d_out, int out_size, void* d_ws, size_t ws_size,
                              hipStream_t stream)
{
    const float* x      = (const float*)d_in[0];   // (2,2048,1024)
    const float* W_in   = (const float*)d_in[1];   // (1024,4096)
    const float* conv_w = (const float*)d_in[2];   // (2048,1,4)
    const float* conv_b = (const float*)d_in[3];   // (2048,)
    const float* W_x    = (const float*)d_in[4];   // (2048,33)
    const float* A_log  = (const float*)d_in[5];   // (2048,16)
    const float* D_par  = (const float*)d_in[6];   // (2048,)
    const float* W_out  = (const float*)d_in[7];   // (2048,1024)
    float* out = (float*)d_out;                    // (2,2048,1024)

    // workspace layout (f32):
    float* xr    = (float*)d_ws;                         // 4096 x 4096  (x_inner | res)
    float* xc    = xr    + (size_t)NTOK * 2 * DINNER;    // 4096 x 2048
    float* yv    = xc    + (size_t)NTOK * DINNER;        // 4096 x 2048
    float* delta = yv    + (size_t)NTOK * DINNER;        // 4096
    float* Bc    = delta + NTOK;                         // 4096 x 16
    float* Cc    = Bc    + (size_t)NTOK * DSTATE;        // 4096 x 16

    // 1) xr = x @ W_in        (M=4096, N=4096, K=1024)
    gemm_f32_wmma<<<dim3((2 * DINNER) / BN, NTOK / BM), 256, 0, stream>>>(
        x, W_in, xr, NTOK, 2 * DINNER, DMODEL);

    // 2) depthwise causal conv + bias + SiLU -> xc
    {
        const size_t n = (size_t)NTOK * DINNER;
        conv_silu_kernel<<<(unsigned)((n + 255) / 256), 256, 0, stream>>>(
            xr, conv_w, conv_b, xc);
    }

    // 3) ssm_params = xc @ W_x ; split into delta/B/C
    ssm_params_kernel<<<NTOK, 256, 0, stream>>>(xc, W_x, delta, Bc, Cc);

    // 4) selective scan -> yv
    scan_kernel<<<BATCH * (DINNER / 256), 256, 0, stream>>>(
        xc, delta, Bc, Cc, A_log, yv);

    // 5) epilogue: yv = (yv + xc*D) * silu(res)
    {
        const size_t n = (size_t)NTOK * DINNER;
        epilogue_kernel<<<(unsigned)((n + 255) / 256), 256, 0, stream>>>(
            yv, xc, xr, D_par);
    }

    // 6) out = yv @ W_out      (M=4096, N=1024, K=2048)
    gemm_f32_wmma<<<dim3(DMODEL / BN, NTOK / BM), 256, 0, stream>>>(
        yv, W_out, out, NTOK, DMODEL, DINNER);
}